// LatentLayer_88441966559693
// MI455X (gfx1250) — compile-verified
//
#include <hip/hip_runtime.h>

// Problem constants (match reference)
#define N_ROWS 16384
#define D_DIM  128
#define M_COLS 4096

// GEMM+LSE tiling
#define BM 64                       // columns per workgroup tile
#define BN 32                       // rows per inner tile (2 waves of 16)
#define ROW_CHUNKS 64               // gridDim.y
#define ROWS_PER_CHUNK (N_ROWS / ROW_CHUNKS)   // 256
#define ITERS (ROWS_PER_CHUNK / BN)            // 8
#define APAD 132                    // padded LDS row stride (floats): 16B-aligned, bank-spread

typedef float v2f __attribute__((ext_vector_type(2)));
typedef float v8f __attribute__((ext_vector_type(8)));

// ---------------------------------------------------------------------------
// z = z_mean + exp(0.5*logvar) * eps   (vectorized float4, exact-size grid)
// ---------------------------------------------------------------------------
__global__ void z_sample_kernel(const float* __restrict__ zm,
                                const float* __restrict__ lv,
                                const float* __restrict__ eps,
                                float* __restrict__ z) {
    int i = blockIdx.x * blockDim.x + threadIdx.x;      // one float4 per thread
    float sigma = __expf(0.5f * lv[0]);
    float4 a = ((const float4*)zm)[i];
    float4 b = ((const float4*)eps)[i];
    float4 r;
    r.x = a.x + sigma * b.x;
    r.y = a.y + sigma * b.y;
    r.z = a.z + sigma * b.z;
    r.w = a.w + sigma * b.w;
    ((float4*)z)[i] = r;
}

// ---------------------------------------------------------------------------
// out[row] = sum_d X[row,d]^2    (one wave32 per row, D=128 -> float4/lane)
// ---------------------------------------------------------------------------
__global__ void row_sumsq_kernel(const float* __restrict__ X,
                                 float* __restrict__ out) {
    int row  = (blockIdx.x * blockDim.x + threadIdx.x) >> 5;
    int lane = threadIdx.x & 31;
    float4 v = ((const float4*)(X + (size_t)row * D_DIM))[lane];
    float s = v.x * v.x + v.y * v.y + v.z * v.z + v.w * v.w;
#pragma unroll
    for (int off = 16; off > 0; off >>= 1) s += __shfl_xor(s, off, 32);
    if (lane == 0) out[row] = s;
}

// ---------------------------------------------------------------------------
// Fused GEMM (z_mean @ e^T) + online logsumexp partials over N.
// Workgroup: 256 threads = 8 waves; tile BN x BM = 32 x 64 (2x4 waves of 16x16)
// B operand (loop-invariant) is preloaded to 64 VGPRs/lane; the hot loop only
// streams A from LDS -> half the LDS traffic per WMMA. Two accumulator chains
// break the serial WMMA C-dependency.
// ---------------------------------------------------------------------------
__global__ __launch_bounds__(256)
void gemm_lse_partial_kernel(const float* __restrict__ zm,
                             const float* __restrict__ e,
                             const float* __restrict__ zz,
                             const float* __restrict__ ee,
                             const float* __restrict__ lv,
                             float* __restrict__ pmax,
                             float* __restrict__ psum) {
    __shared__ float As[BN * APAD];
    __shared__ float Bs[BM * APAD];
    __shared__ float redM[4][16];
    __shared__ float redP[4][16];

    const int tid  = threadIdx.x;
    const int lane = tid & 31;
    const int wave = tid >> 5;
    const int wr   = wave & 1;   // which 16-row half of the BN tile
    const int wc   = wave >> 1;  // which 16-col tile (0..3)
    const int colBase  = blockIdx.x * BM;
    const int rowChunk = blockIdx.y * ROWS_PER_CHUNK;

    const float alpha = -0.5f * __expf(lv[0]);   // -0.5 * exp(logvar)

    // Stage B tile once: e[colBase .. colBase+63][0..127] (invariant over it-loop)
    {
        const float4* src = (const float4*)e;
#pragma unroll
        for (int i = 0; i < (BM * D_DIM / 4) / 256; ++i) {   // 8 float4 per thread
            int idx = tid + i * 256;
            int r = idx >> 5;          // 32 float4 per row
            int c = idx & 31;
            float4 v = src[(size_t)(colBase + r) * (D_DIM / 4) + c];
            *(float4*)&Bs[r * APAD + c * 4] = v;
        }
    }

    // WMMA f32 16x16x4 lane mapping: lanes 0-15 hold K={0,1}, lanes 16-31 K={2,3}
    const int rowA = lane & 15;
    const int kg   = (lane >> 4) << 1;
    const int colg = colBase + wc * 16 + rowA;           // valid for lanes < 16
    const float eecol = ee[colBase + wc * 16 + rowA];

    __syncthreads();   // Bs ready

    // Preload this wave's whole B operand (64 floats/lane = 64 VGPRs)
    v2f breg[32];
    {
        const float* Bb = &Bs[(wc * 16 + rowA) * APAD + kg];
#pragma unroll
        for (int i = 0; i < 32; ++i) breg[i] = *(const v2f*)(Bb + 4 * i);
    }

    float runM = -INFINITY;
    float runP = 0.0f;

    for (int it = 0; it < ITERS; ++it) {
        __syncthreads();   // previous compute done (and breg preload at it==0)
        // Stage A tile: z_mean[rowChunk + it*BN .. +31][0..127]
        {
            const float4* src = (const float4*)zm;
            int rbase = rowChunk + it * BN;
#pragma unroll
            for (int i = 0; i < (BN * D_DIM / 4) / 256; ++i) {  // 4 float4 per thread
                int idx = tid + i * 256;
                int r = idx >> 5;
                int c = idx & 31;
                float4 v = src[(size_t)(rbase + r) * (D_DIM / 4) + c];
                *(float4*)&As[r * APAD + c * 4] = v;
            }
            // Prefetch next A tile toward the WGP (global_prefetch_b8)
            if (it + 1 < ITERS) {
                const float* nxt = zm + (size_t)(rowChunk + (it + 1) * BN + (tid >> 3)) * D_DIM
                                      + (tid & 7) * 16;
                __builtin_prefetch(nxt, 0, 1);
            }
        }
        __syncthreads();   // As ready

        // 16x16 tile of z_mean @ e^T: 32 x V_WMMA_F32_16X16X4_F32, 2 acc chains
        v8f acc0 = {};
        v8f acc1 = {};
        const float* Ab = &As[(wr * 16 + rowA) * APAD + kg];
#pragma unroll
        for (int i = 0; i < 32; i += 2) {
            v2f a0 = *(const v2f*)(Ab + 4 * i);
            v2f a1 = *(const v2f*)(Ab + 4 * (i + 1));
            acc0 = __builtin_amdgcn_wmma_f32_16x16x4_f32(
                false, a0, false, breg[i], (short)0, acc0, false, false);
            acc1 = __builtin_amdgcn_wmma_f32_16x16x4_f32(
                false, a1, false, breg[i + 1], (short)0, acc1, false, false);
        }
        v8f acc = acc0 + acc1;

        // s = alpha * (zz_i + ee_j - 2*dot); online (max, sumexp) over 16 rows
        int rbaseg = rowChunk + it * BN + wr * 16 + ((lane >> 4) << 3);
        float s[8];
        float mt = -INFINITY;
#pragma unroll
        for (int r = 0; r < 8; ++r) {
            float dist = zz[rbaseg + r] + eecol - 2.0f * acc[r];
            s[r] = alpha * dist;
            mt = fmaxf(mt, s[r]);
        }
        float pt = 0.0f;
#pragma unroll
        for (int r = 0; r < 8; ++r) pt += __expf(s[r] - mt);

        // merge row-halves held in lane pairs (l, l^16)
        float mo = __shfl_xor(mt, 16, 32);
        float po = __shfl_xor(pt, 16, 32);
        float nm = fmaxf(mt, mo);
        pt = pt * __expf(mt - nm) + po * __expf(mo - nm);
        mt = nm;

        // fold into running accumulator across row-tiles
        nm   = fmaxf(runM, mt);
        runP = runP * __expf(runM - nm) + pt * __expf(mt - nm);
        runM = nm;
    }

    __syncthreads();
    if (wr == 0 && lane < 16) { redM[wc][lane] = runM; redP[wc][lane] = runP; }
    __syncthreads();
    if (wr == 1 && lane < 16) {
        float m2 = redM[wc][lane];
        float p2 = redP[wc][lane];
        float nm = fmaxf(runM, m2);
        float pp = runP * __expf(runM - nm) + p2 * __expf(m2 - nm);
        size_t o = (size_t)blockIdx.y * M_COLS + colg;
        pmax[o] = nm;
        psum[o] = pp;
    }
}

// ---------------------------------------------------------------------------
// Merge per-chunk partials -> lse per column -> mean -> final scalar loss.
// Single workgroup; tiny amount of work (64*4096 pairs).
// ---------------------------------------------------------------------------
__global__ void reduce_loss_kernel(const float* __restrict__ pmax,
                                   const float* __restrict__ psum,
                                   const float* __restrict__ lv,
                                   float* __restrict__ loss) {
    __shared__ float sred[256];
    int tid = threadIdx.x;
    float accv = 0.0f;
    for (int j = tid; j < M_COLS; j += 256) {
        float m = -INFINITY;
        for (int k = 0; k < ROW_CHUNKS; ++k)
            m = fmaxf(m, pmax[(size_t)k * M_COLS + j]);
        float S = 0.0f;
        for (int k = 0; k < ROW_CHUNKS; ++k)
            S += psum[(size_t)k * M_COLS + j] * __expf(pmax[(size_t)k * M_COLS + j] - m);
        accv += -(m + __logf(S));
    }
    sred[tid] = accv;
    __syncthreads();
    for (int s2 = 128; s2 > 0; s2 >>= 1) {
        if (tid < s2) sred[tid] += sred[tid + s2];
        __syncthreads();
    }
    if (tid == 0)
        loss[0] = sred[0] / (float)M_COLS + 0.5f * (float)D_DIM * lv[0];
}

// ---------------------------------------------------------------------------
extern "C" void kernel_launch(void* const* d_in, const int* in_sizes, int n_in,
                              void* d_out, int out_size, void* d_ws, size_t ws_size,
                              hipStream_t stream) {
    (void)in_sizes; (void)n_in; (void)out_size; (void)ws_size;

    const float* z_mean   = (const float*)d_in[0];
    const float* z_logvar = (const float*)d_in[1];
    const float* eps      = (const float*)d_in[2];
    const float* e        = (const float*)d_in[3];
    // d_in[4] is M (int scalar) -- compile-time constant here.

    float* z_out    = (float*)d_out;                     // [N, D]
    float* loss_out = z_out + (size_t)N_ROWS * D_DIM;    // scalar

    // Workspace layout (floats): zz[N] | ee[M] | pmax[64*M] | psum[64*M]  (~2.2 MB)
    float* zz   = (float*)d_ws;
    float* ee   = zz + N_ROWS;
    float* pmax = ee + M_COLS;
    float* psum = pmax + (size_t)ROW_CHUNKS * M_COLS;

    // 1) z sample (2M floats, float4-vectorized; exact grid)
    z_sample_kernel<<<(N_ROWS * D_DIM / 4) / 256, 256, 0, stream>>>(
        z_mean, z_logvar, eps, z_out);

    // 2) row squared norms (one wave per row)
    row_sumsq_kernel<<<(N_ROWS * 32) / 256, 256, 0, stream>>>(z_mean, zz);
    row_sumsq_kernel<<<(M_COLS * 32) / 256, 256, 0, stream>>>(e, ee);

    // 3) fused GEMM + online logsumexp partials
    dim3 grid(M_COLS / BM, ROW_CHUNKS);   // (64, 64)
    gemm_lse_partial_kernel<<<grid, 256, 0, stream>>>(
        z_mean, e, zz, ee, z_logvar, pmax, psum);

    // 4) final merge + mean + scalar term
    reduce_loss_kernel<<<1, 256, 0, stream>>>(pmax, psum, z_logvar, loss_out);
}